// Model_59931973648387
// MI455X (gfx1250) — compile-verified
//
#include <hip/hip_runtime.h>
#include <hip/hip_bf16.h>

typedef __attribute__((ext_vector_type(2))) float v2f;
typedef __attribute__((ext_vector_type(8))) float v8f;

static constexpr int TLEN = 512;
static constexpr int NB   = 256;

// Branch-free tanh: native V_TANH_F32 on gfx1250 when the builtin exists,
// otherwise 2 transcendentals (exp2 + rcp) with no EXEC divergence.
__device__ __forceinline__ float fast_tanh(float x)
{
#if __has_builtin(__builtin_amdgcn_tanhf)
    return __builtin_amdgcn_tanhf(x);
#else
    const float ax = fabsf(x);
#if __has_builtin(__builtin_amdgcn_exp2f)
    const float e = __builtin_amdgcn_exp2f(ax * -2.885390081777927f); // exp(-2|x|)
#else
    const float e = __expf(-2.0f * ax);
#endif
#if __has_builtin(__builtin_amdgcn_rcpf)
    const float r = (1.0f - e) * __builtin_amdgcn_rcpf(1.0f + e);
#else
    const float r = (1.0f - e) / (1.0f + e);
#endif
    return copysignf(r, x);
#endif
}

// One wave32 = one 16-batch tile. State kept as Z^T (h^T) in WMMA C/D layout:
//   lane: batch column n = lane&15 ; VGPR v: h row = v + 8*(lane>>4)
// Each step: load all x_t chunks (one clause); acc = bias; NC input WMMAs;
//            4 recurrent WMMAs (B built from state via lane-half swap);
//            h = act(acc); store h row-major.
// ACT: 0 = tanh, 1 = relu (compile-time, keeps the hot loop branch-free).
template<int DIN, int ACT>
__global__ __launch_bounds__(32)
void rnn_seq_kernel(const float* __restrict__ in, long long in_rst, long long in_rsb,
                    const float* __restrict__ Wih, const float* __restrict__ Whh,
                    const float* __restrict__ bih, const float* __restrict__ bhh,
                    const float* __restrict__ h0,
                    float* __restrict__ out, long long o_rst, long long o_rsb, int ooff,
                    float* __restrict__ hT_out, int reverse)
{
    constexpr int NC = (DIN + 3) / 4;     // K chunks of 4 for the input projection
    const int lane = threadIdx.x;
    const int bloc = lane & 15;           // batch-in-tile / matrix row m
    const int hf   = lane >> 4;           // lane half
    const int B    = blockIdx.x * 16 + bloc;

    // ---- preload weights as WMMA A operands (16x4 f32: m = lane&15, K = 4c+2*hf+{0,1})
    v2f aih[NC];
#pragma unroll
    for (int c = 0; c < NC; ++c) {
        const int k0 = 4 * c + 2 * hf;
        aih[c][0] = (k0     < DIN) ? Wih[bloc * DIN + k0    ] : 0.0f;
        aih[c][1] = (k0 + 1 < DIN) ? Wih[bloc * DIN + k0 + 1] : 0.0f;
    }
    v2f ahh[4];
#pragma unroll
    for (int c = 0; c < 4; ++c) {
        const int k0 = 4 * c + 2 * hf;
        ahh[c][0] = Whh[bloc * 16 + k0];
        ahh[c][1] = Whh[bloc * 16 + k0 + 1];
    }
    float bias[8];
#pragma unroll
    for (int v = 0; v < 8; ++v) {
        const int hh = 8 * hf + v;
        float b = 0.0f;
        if (bih) b += bih[hh];
        if (bhh) b += bhh[hh];
        bias[v] = b;
    }

    // ---- initial hidden state (natural [N,16] layout in memory)
    v8f h;
    if (h0) {
        const float* hp = h0 + (size_t)B * 16 + 8 * hf;
#pragma unroll
        for (int v = 0; v < 8; ++v) h[v] = hp[v];
    } else {
#pragma unroll
        for (int v = 0; v < 8; ++v) h[v] = 0.0f;
    }

    // ---- pointer-stepped sequence addressing (uniform signed strides)
    const long long t0    = reverse ? (long long)(TLEN - 1) : 0LL;
    const long long istep = reverse ? -in_rst : in_rst;
    const long long ostep = reverse ? -o_rst  : o_rst;
    const float* ip = in  + t0 * in_rst + (long long)B * in_rsb;
    float*       op = out + t0 * o_rst  + (long long)B * o_rsb + ooff + 8 * hf;

    for (int i = 0; i < TLEN; ++i) {
        // ---- load all input chunks first (distinct registers -> one load clause)
        v2f xb[NC];
#pragma unroll
        for (int c = 0; c < NC; ++c) {
            const int k0 = 4 * c + 2 * hf;
            if constexpr ((DIN % 8) == 0) {
                const float2 f = *(const float2*)(ip + k0);   // 8B aligned
                xb[c][0] = f.x; xb[c][1] = f.y;
            } else {
                xb[c][0] = (k0     < DIN) ? ip[k0    ] : 0.0f;
                xb[c][1] = (k0 + 1 < DIN) ? ip[k0 + 1] : 0.0f;
            }
        }

        // ---- build recurrent B chunks from the D-layout state (lane-half swap)
        const float s0 = __shfl_xor(h[0], 16, 32);
        const float s1 = __shfl_xor(h[1], 16, 32);
        const float s2 = __shfl_xor(h[2], 16, 32);
        const float s3 = __shfl_xor(h[3], 16, 32);
        const float s4 = __shfl_xor(h[4], 16, 32);
        const float s5 = __shfl_xor(h[5], 16, 32);
        const float s6 = __shfl_xor(h[6], 16, 32);
        const float s7 = __shfl_xor(h[7], 16, 32);
        v2f r0, r1, r2, r3;
        r0[0] = hf ? s2   : h[0]; r0[1] = hf ? s3   : h[1];  // K = 0..3
        r1[0] = hf ? s6   : h[4]; r1[1] = hf ? s7   : h[5];  // K = 4..7
        r2[0] = hf ? h[2] : s0;   r2[1] = hf ? h[3] : s1;    // K = 8..11
        r3[0] = hf ? h[6] : s4;   r3[1] = hf ? h[7] : s5;    // K = 12..15

        // ---- WMMA chain: bias + input projection + recurrent term
        v8f acc;
#pragma unroll
        for (int v = 0; v < 8; ++v) acc[v] = bias[v];
#pragma unroll
        for (int c = 0; c < NC; ++c)
            acc = __builtin_amdgcn_wmma_f32_16x16x4_f32(false, aih[c], false, xb[c],
                                                        (short)0, acc, false, false);
        acc = __builtin_amdgcn_wmma_f32_16x16x4_f32(false, ahh[0], false, r0, (short)0, acc, false, false);
        acc = __builtin_amdgcn_wmma_f32_16x16x4_f32(false, ahh[1], false, r1, (short)0, acc, false, false);
        acc = __builtin_amdgcn_wmma_f32_16x16x4_f32(false, ahh[2], false, r2, (short)0, acc, false, false);
        acc = __builtin_amdgcn_wmma_f32_16x16x4_f32(false, ahh[3], false, r3, (short)0, acc, false, false);

        // ---- activation (branch-free, wave stays fully converged)
#pragma unroll
        for (int v = 0; v < 8; ++v) {
            const float z = acc[v];
            if constexpr (ACT == 1) h[v] = fmaxf(z, 0.0f);
            else                    h[v] = fast_tanh(z);
        }

        // ---- store y_t row-major (lane covers h = 8*hf .. 8*hf+7, 32B aligned)
        *(float4*)(op)     = make_float4(h[0], h[1], h[2], h[3]);
        *(float4*)(op + 4) = make_float4(h[4], h[5], h[6], h[7]);

        ip += istep;
        op += ostep;
    }

    if (hT_out) {
        float* hp = hT_out + (size_t)B * 16 + 8 * hf;
        *(float4*)(hp)     = make_float4(h[0], h[1], h[2], h[3]);
        *(float4*)(hp + 4) = make_float4(h[4], h[5], h[6], h[7]);
    }
}

// ---------------------------------------------------------------------------

static void run_layer(hipStream_t s, const float* in, long long irst, long long irsb, int Din,
                      const float* Wih, const float* Whh, const float* bih, const float* bhh,
                      const float* h0, float* out, long long orst, long long orsb, int ooff,
                      float* hT, int rev, int act)
{
    dim3 g(16), b(32);
#define LAUNCH(DN, AC) rnn_seq_kernel<DN, AC><<<g, b, 0, s>>>(in, irst, irsb, Wih, Whh, bih, bhh, h0, \
                                                              out, orst, orsb, ooff, hT, rev)
    if (act == 1) {
        switch (Din) { case 16: LAUNCH(16, 1); break; case 25: LAUNCH(25, 1); break; default: LAUNCH(32, 1); break; }
    } else {
        switch (Din) { case 16: LAUNCH(16, 0); break; case 25: LAUNCH(25, 0); break; default: LAUNCH(32, 0); break; }
    }
#undef LAUNCH
}

// 4-layer bidirectional run (PyTorch nn.RNN semantics), ping-pong through PP0/PP1.
static void bidir_run(hipStream_t s, const float* const* w, const float* in16,
                      float* outF, long long o_rst, long long o_rsb,
                      const float* h0base, float* hTbase, int act,
                      float* PP0, float* PP1)
{
    for (int l = 0; l < 4; ++l) {
        const float* lin; long long irst, irsb; int Din;
        if (l == 0) { lin = in16; irst = (long long)NB * 16; irsb = 16; Din = 16; }
        else        { lin = (l & 1) ? PP0 : PP1; irst = (long long)NB * 32; irsb = 32; Din = 32; }
        float* lout; long long orst, orsb;
        if (l == 3) { lout = outF; orst = o_rst; orsb = o_rsb; }
        else        { lout = (l & 1) ? PP1 : PP0; orst = (long long)NB * 32; orsb = 32; }
        for (int d = 0; d < 2; ++d) {
            const float* const* wd = w + (l * 2 + d) * 4;
            const float* h0 = h0base ? (h0base + (size_t)(2 * l + d) * NB * 16) : nullptr;
            float*       hT = hTbase ? (hTbase + (size_t)(2 * l + d) * NB * 16) : nullptr;
            run_layer(s, lin, irst, irsb, Din, wd[0], wd[1], wd[2], wd[3], h0,
                      lout, orst, orsb, d ? 16 : 0, hT, d, act);
        }
    }
}

extern "C" void kernel_launch(void* const* d_in, const int* in_sizes, int n_in,
                              void* d_out_v, int out_size, void* d_ws, size_t ws_size,
                              hipStream_t stream)
{
    (void)in_sizes; (void)out_size; (void)ws_size;
    const float* prm[160];
    for (int i = 0; i < n_in && i < 160; ++i) prm[i] = (const float*)d_in[i];

    int ip = 0;
    const float* x = prm[ip++];
    const float* y = prm[ip++];
    const float* p00[4];  for (int i = 0; i < 4;  ++i) p00[i] = prm[ip++];
    const float* p01[6];  for (int i = 0; i < 6;  ++i) p01[i] = prm[ip++];
    const float* p02[32]; for (int i = 0; i < 32; ++i) p02[i] = prm[ip++];
    const float* p03[32]; for (int i = 0; i < 32; ++i) p03[i] = prm[ip++];
    const float* p10[4];  for (int i = 0; i < 4;  ++i) p10[i] = prm[ip++];
    const float* p11[6];  for (int i = 0; i < 6;  ++i) p11[i] = prm[ip++];
    const float* p12[32]; for (int i = 0; i < 32; ++i) p12[i] = prm[ip++];
    const float* p13[32]; for (int i = 0; i < 32; ++i) p13[i] = prm[ip++];

    float* out = (float*)d_out_v;
    const long long SEQ16 = (long long)TLEN * NB * 16;   // 2,097,152 floats
    const long long SEQ32 = (long long)TLEN * NB * 32;   // 4,194,304 floats

    // workspace: two [T,N,16] and two [T,N,32] sequence buffers (~50 MB)
    float* S16a = (float*)d_ws;
    float* S16b = S16a + SEQ16;
    float* PP0  = S16b + SEQ16;
    float* PP1  = PP0 + SEQ32;

    // d_out layout (reference return order, flattened)
    float* o_x2 = out;
    float* o_x3 = o_x2 + SEQ32;
    float* o_h0 = o_x3 + SEQ32;
    float* o_h2 = o_h0 + (long long)NB * 16;
    float* o_h3 = o_h2 + 8LL * NB * 16;
    float* o_y2 = o_h3 + 8LL * NB * 16;
    float* o_y3 = o_y2 + SEQ32;
    float* o_h4 = o_y3 + SEQ32;
    float* o_h6 = o_h4 + (long long)NB * 16;
    float* o_h7 = o_h6 + 8LL * NB * 16;

    const int TANH = 0, RELU = 1;

    // ---------------- x path: x is [T,N,32]
    run_layer(stream, x, (long long)NB * 32, 32, 32, p00[0], p00[1], p00[2], p00[3], nullptr,
              S16a, (long long)NB * 16, 16, 0, o_h0, 0, TANH);
    // p01: 3 uni layers, no bias, Din=16: S16a -> PP0 -> PP1 -> S16b (=x1)
    run_layer(stream, S16a, (long long)NB * 16, 16, 16, p01[0], p01[1], nullptr, nullptr, nullptr,
              PP0, (long long)NB * 16, 16, 0, nullptr, 0, TANH);
    run_layer(stream, PP0, (long long)NB * 16, 16, 16, p01[2], p01[3], nullptr, nullptr, nullptr,
              PP1, (long long)NB * 16, 16, 0, nullptr, 0, TANH);
    run_layer(stream, PP1, (long long)NB * 16, 16, 16, p01[4], p01[5], nullptr, nullptr, nullptr,
              S16b, (long long)NB * 16, 16, 0, nullptr, 0, TANH);
    // p02: relu bidir, out x2 [T,N,32], finals -> h2
    bidir_run(stream, p02, S16b, o_x2, (long long)NB * 32, 32, nullptr, o_h2, RELU, PP0, PP1);
    // p03: tanh bidir with h0 = h2, out x3, finals -> h3
    bidir_run(stream, p03, S16b, o_x3, (long long)NB * 32, 32, o_h2, o_h3, TANH, PP0, PP1);

    // ---------------- y path: y is batch-first [N,T,25] -> t-stride 25, b-stride T*25
    run_layer(stream, y, 25, (long long)TLEN * 25, 25, p10[0], p10[1], p10[2], p10[3], nullptr,
              S16a, (long long)NB * 16, 16, 0, o_h4, 0, TANH);
    run_layer(stream, S16a, (long long)NB * 16, 16, 16, p11[0], p11[1], nullptr, nullptr, nullptr,
              PP0, (long long)NB * 16, 16, 0, nullptr, 0, TANH);
    run_layer(stream, PP0, (long long)NB * 16, 16, 16, p11[2], p11[3], nullptr, nullptr, nullptr,
              PP1, (long long)NB * 16, 16, 0, nullptr, 0, TANH);
    run_layer(stream, PP1, (long long)NB * 16, 16, 16, p11[4], p11[5], nullptr, nullptr, nullptr,
              S16b, (long long)NB * 16, 16, 0, nullptr, 0, TANH);
    // final y layers store directly in swapped [N,T,32] layout: t-stride 32, b-stride T*32
    bidir_run(stream, p12, S16b, o_y2, 32, (long long)TLEN * 32, nullptr, o_h6, RELU, PP0, PP1);
    bidir_run(stream, p13, S16b, o_y3, 32, (long long)TLEN * 32, o_h6, o_h7, TANH, PP0, PP1);
}